// LightSBM_1262720385516
// MI455X (gfx1250) — compile-verified
//
#include <hip/hip_runtime.h>
#include <math.h>

// ---------------------------------------------------------------------------
// LightSBM fused kernel for gfx1250 (MI455X).
//   logits[b,k] = sum_d 0.5*x^2*S[k,d] + sum_d x*r[k,d] + log_alpha[k]
//   idx = argmax_k(logits + gumbel);  out = r[idx] + S[idx]*x + sqrt(S[idx])*noise
// GEMM done with V_WMMA_F32_16X16X4_F32 (exact fp32), inner dim 256 (augmented).
// ---------------------------------------------------------------------------

typedef float v2f __attribute__((ext_vector_type(2)));
typedef float v4f __attribute__((ext_vector_type(4)));
typedef float v8f __attribute__((ext_vector_type(8)));

#define BATCH   262144
#define DIM     128
#define NK      128
#define TM      64          // rows per workgroup
#define XPITCH  132         // LDS pitch in floats (conflict-free A-fragment reads)

// ---------------------------------------------------------------------------
// Prep: Sexp = exp(S_log_diag)  and the WMMA-swizzled augmented weight blob.
// Wf float index f = ((tile*64 + s)*32 + lane)*2 + q
//   tile in [0,8): N-tile (16 cols);  s in [0,64): inner step (k0 = 4s)
//   lane in [0,32), q in {0,1}:  kk = 4s + (lane/16)*2 + q,  n = tile*16 + lane%16
//   value = kk < 128 ? 0.5*exp(S_log_diag[n, kk]) : r[n, kk-128]
// ---------------------------------------------------------------------------
__global__ __launch_bounds__(256) void lightsbm_prep(
    const float* __restrict__ S_log_diag,
    const float* __restrict__ r,
    float* __restrict__ Sexp,
    float* __restrict__ Wf)
{
    const int f = blockIdx.x * 256 + threadIdx.x;   // 0 .. 32767
    if (f < NK * DIM) {
        Sexp[f] = __expf(S_log_diag[f]);
    }
    const int tile = f >> 12;          // /4096
    const int rem  = f & 4095;
    const int s    = rem >> 6;
    const int l2   = rem & 63;
    const int lane = l2 >> 1;
    const int q    = l2 & 1;
    const int kk   = 4 * s + ((lane >> 4) << 1) + q;   // augmented inner index
    const int n    = tile * 16 + (lane & 15);          // mixture component
    float v;
    if (kk < DIM) v = 0.5f * __expf(S_log_diag[n * DIM + kk]);
    else          v = r[n * DIM + (kk - DIM)];
    Wf[f] = v;
}

// ---------------------------------------------------------------------------
// Fused GEMM + gumbel-argmax + gather + reparameterized sample.
// grid = B/64 workgroups, 256 threads (8 wave32).
// wave w: tile-row (w&3) [16 rows], n-half (w>>2) [4 tiles of 16 cols].
// ---------------------------------------------------------------------------
__global__ __launch_bounds__(256, 2) void lightsbm_main(
    const float* __restrict__ x,
    const float* __restrict__ log_alpha,
    const float* __restrict__ r,
    const float* __restrict__ gumbel,
    const float* __restrict__ noise,
    const float* __restrict__ Sexp,
    const float* __restrict__ Wf,
    float* __restrict__ out)
{
    __shared__ float xs[TM * XPITCH];     // raw x tile, 33.8 KB
    __shared__ float candv[2][TM];
    __shared__ int   candi[2][TM];

    const int t    = threadIdx.x;
    const int wave = t >> 5;
    const int lane = t & 31;
    const int rowBase = blockIdx.x * TM;

    // ---- stage x tile into LDS (coalesced float4) ----
    #pragma unroll
    for (int i = 0; i < 8; ++i) {
        const int e4 = i * 256 + t;           // 0..2047 float4s
        const int rl = e4 >> 5;               // local row 0..63
        const int c4 = (e4 & 31) << 2;        // col 0..124 step 4
        v4f xv = *(const v4f*)(x + (rowBase + rl) * DIM + c4);
        *(v4f*)(&xs[rl * XPITCH + c4]) = xv;
    }
    __syncthreads();

    const int tr = wave & 3;          // tile-row within WG
    const int nh = wave >> 2;         // column half (0/1)
    const int m0 = tr * 16;
    const int hs = lane >> 4;         // half-lane select
    const int lm = lane & 15;

    const v8f vzero = {0.f, 0.f, 0.f, 0.f, 0.f, 0.f, 0.f, 0.f};
    v8f acc[4] = {vzero, vzero, vzero, vzero};

    // A-fragment base: row m0+lm, col offset 2*hs (lanes 16-31 take K+2,K+3)
    const float* xrow = &xs[(m0 + lm) * XPITCH + 2 * hs];
    // B-fragment base for this wave's 4 tiles
    const float* wf0 = Wf + ((nh * 4) * 64) * 64 + lane * 2;

    // ---- inner dim 0..127: squared term (0.5*x^2 @ S) ----
    #pragma unroll 4
    for (int s = 0; s < 32; ++s) {
        v2f a = *(const v2f*)(xrow + 4 * s);
        a.x *= a.x; a.y *= a.y;
        #pragma unroll
        for (int tt = 0; tt < 4; ++tt) {
            v2f b = *(const v2f*)(wf0 + (tt * 64 + s) * 64);
            acc[tt] = __builtin_amdgcn_wmma_f32_16x16x4_f32(
                false, a, false, b, (short)0, acc[tt], false, false);
        }
    }
    // ---- inner dim 128..255: linear term (x @ r) ----
    #pragma unroll 4
    for (int s = 32; s < 64; ++s) {
        v2f a = *(const v2f*)(xrow + 4 * s - DIM);
        #pragma unroll
        for (int tt = 0; tt < 4; ++tt) {
            v2f b = *(const v2f*)(wf0 + (tt * 64 + s) * 64);
            acc[tt] = __builtin_amdgcn_wmma_f32_16x16x4_f32(
                false, a, false, b, (short)0, acc[tt], false, false);
        }
    }

    // ---- add log_alpha + gumbel, per-row argmax over this wave's 64 cols ----
    float la[4];
    #pragma unroll
    for (int tt = 0; tt < 4; ++tt)
        la[tt] = log_alpha[(nh * 4 + tt) * 16 + lm];

    #pragma unroll
    for (int j = 0; j < 8; ++j) {
        // C/D layout: VGPR j, lanes 0-15 -> row m0+j, lanes 16-31 -> row m0+j+8
        const int rl   = m0 + j + 8 * hs;
        const int grow = rowBase + rl;
        float bestv = -INFINITY;
        int   bestk = 0;
        #pragma unroll
        for (int tt = 0; tt < 4; ++tt) {
            const int col = (nh * 4 + tt) * 16 + lm;
            const float v = acc[tt][j] + la[tt] + gumbel[grow * NK + col];
            if (v > bestv) { bestv = v; bestk = col; }
        }
        // butterfly reduce across the 16 lanes holding this row
        #pragma unroll
        for (int off = 8; off > 0; off >>= 1) {
            const float ov = __shfl_xor(bestv, off, 16);
            const int   ok = __shfl_xor(bestk, off, 16);
            if (ov > bestv || (ov == bestv && ok < bestk)) { bestv = ov; bestk = ok; }
        }
        if (lm == 0) { candv[nh][rl] = bestv; candi[nh][rl] = bestk; }
    }
    __syncthreads();

    // ---- gather + reparameterized sample (EPS = 1): coalesced float4 ----
    #pragma unroll
    for (int i = 0; i < 8; ++i) {
        const int e4 = i * 256 + t;
        const int rl = e4 >> 5;
        const int c4 = (e4 & 31) << 2;
        const float v0 = candv[0][rl];
        const float v1 = candv[1][rl];
        const int   k  = (v1 > v0) ? candi[1][rl] : candi[0][rl];
        const int grow = rowBase + rl;

        v4f xv = *(const v4f*)(&xs[rl * XPITCH + c4]);
        v4f nz = *(const v4f*)(noise + grow * DIM + c4);
        v4f Sv = *(const v4f*)(Sexp + k * DIM + c4);
        v4f rv = *(const v4f*)(r + k * DIM + c4);
        v4f o;
        o.x = rv.x + Sv.x * xv.x + __fsqrt_rn(Sv.x) * nz.x;
        o.y = rv.y + Sv.y * xv.y + __fsqrt_rn(Sv.y) * nz.y;
        o.z = rv.z + Sv.z * xv.z + __fsqrt_rn(Sv.z) * nz.z;
        o.w = rv.w + Sv.w * xv.w + __fsqrt_rn(Sv.w) * nz.w;
        *(v4f*)(out + grow * DIM + c4) = o;
    }
}

// ---------------------------------------------------------------------------
extern "C" void kernel_launch(void* const* d_in, const int* in_sizes, int n_in,
                              void* d_out, int out_size, void* d_ws, size_t ws_size,
                              hipStream_t stream) {
    const float* x          = (const float*)d_in[0];   // [B, DIM]
    const float* log_alpha  = (const float*)d_in[1];   // [K]
    const float* r          = (const float*)d_in[2];   // [K, DIM]
    const float* S_log_diag = (const float*)d_in[3];   // [K, DIM]
    const float* gumbel     = (const float*)d_in[4];   // [B, K]
    const float* noise      = (const float*)d_in[5];   // [B, DIM]
    float* out = (float*)d_out;

    // workspace: Sexp (64 KB) | Wf swizzled weights (128 KB)
    float* Sexp = (float*)d_ws;
    float* Wf   = Sexp + NK * DIM;

    lightsbm_prep<<<dim3(128), dim3(256), 0, stream>>>(S_log_diag, r, Sexp, Wf);
    lightsbm_main<<<dim3(BATCH / TM), dim3(256), 0, stream>>>(
        x, log_alpha, r, gumbel, noise, Sexp, Wf, out);
}